// FactorizationMachine_75144747810873
// MI455X (gfx1250) — compile-verified
//
#include <hip/hip_runtime.h>
#include <hip/hip_bf16.h>

// Problem constants (match reference)
#define FM_NNZ   819200
#define FM_BATCH 4096
#define FM_VOCAB 1000000
#define FM_K     16

typedef __attribute__((ext_vector_type(2))) float v2f;
typedef __attribute__((ext_vector_type(8))) float v8f;

// ---------------------------------------------------------------------------
// Kernel 1: zero the accumulators in workspace (S[4096][16] ++ T[4096]).
// Must run every call: harness poisons d_ws once and never re-poisons.
// ---------------------------------------------------------------------------
__global__ void fm_zero_kernel(float* __restrict__ ws, int n) {
  int i = blockIdx.x * blockDim.x + threadIdx.x;
  if (i < n) ws[i] = 0.0f;
}

// ---------------------------------------------------------------------------
// Kernel 2: gather + segmented reduction.
// One wave32 per 32 consecutive nnz (NNZ % 32 == 0 -> every wave is full,
// EXEC all-ones for the shuffles). batch_ids are sorted, so equal ids form
// contiguous runs; a flag-free segmented inclusive scan needs only the
// condition b[lane-d] == b[lane]. Run-end lanes flush with f32 atomics.
// Memory behavior: 4x global_load_b128 per lane for the weight row (random,
// but the 64 MB table is L2-resident on a 192 MB L2), fully coalesced
// b32 loads for indices/batch_ids/values.
// ---------------------------------------------------------------------------
__global__ __launch_bounds__(256) void fm_scatter_kernel(
    const int*   __restrict__ indices,
    const int*   __restrict__ batch_ids,
    const float* __restrict__ values,
    const float* __restrict__ weight,
    float*       __restrict__ S,   // [FM_BATCH][FM_K]
    float*       __restrict__ T)   // [FM_BATCH]
{
  const int i    = blockIdx.x * blockDim.x + threadIdx.x;  // one nnz per lane
  const int lane = threadIdx.x & 31;
  if (i >= FM_NNZ) return;  // never taken with exact launch; waves stay full

  const int   b   = batch_ids[i];
  const int   idx = indices[i];
  const float val = values[i];

  // Gather one K=16 row as 4x float4 (64 B, 16B-aligned since rows are 64 B).
  const float4* wrow = (const float4*)(weight + (size_t)idx * FM_K);
  float4 w0 = wrow[0], w1 = wrow[1], w2 = wrow[2], w3 = wrow[3];

  float vx[FM_K];
  vx[0]=w0.x*val; vx[1]=w0.y*val; vx[2]=w0.z*val; vx[3]=w0.w*val;
  vx[4]=w1.x*val; vx[5]=w1.y*val; vx[6]=w1.z*val; vx[7]=w1.w*val;
  vx[8]=w2.x*val; vx[9]=w2.y*val; vx[10]=w2.z*val; vx[11]=w2.w*val;
  vx[12]=w3.x*val; vx[13]=w3.y*val; vx[14]=w3.z*val; vx[15]=w3.w*val;

  float sq = 0.0f;
#pragma unroll
  for (int k = 0; k < FM_K; ++k) sq += vx[k] * vx[k];

  // Segmented inclusive scan across the wave (17 values, 5 steps).
#pragma unroll
  for (int d = 1; d < 32; d <<= 1) {
    int  bo   = __shfl_up(b, d, 32);
    bool take = (lane >= d) && (bo == b);
#pragma unroll
    for (int k = 0; k < FM_K; ++k) {
      float o = __shfl_up(vx[k], d, 32);
      if (take) vx[k] += o;
    }
    float osq = __shfl_up(sq, d, 32);
    if (take) sq += osq;
  }

  // Run-end lanes hold the full sum of their run inside this window.
  int  bnext = __shfl_down(b, 1, 32);
  bool isEnd = (lane == 31) || (bnext != b);
  if (isEnd) {
#pragma unroll
    for (int k = 0; k < FM_K; ++k)
      atomicAdd(&S[b * FM_K + k], vx[k]);
    atomicAdd(&T[b], sq);
  }
}

// ---------------------------------------------------------------------------
// Kernel 3: out[b] = diag(S * S^T)[b] - T[b], via V_WMMA_F32_16X16X4_F32.
// One wave32 per 16 batches. For chunk c (K-slice 4c..4c+3):
//   A (16x4 f32, 2 VGPRs): lanes 0-15 hold M=lane, K={4c,4c+1};
//                          lanes 16-31 hold M=lane-16, K={4c+2,4c+3}.
//   B (4x16 f32, 2 VGPRs): lanes 0-15 hold K={4c,4c+1}, N=lane;
//                          lanes 16-31 hold K={4c+2,4c+3}, N=lane-16.
// Since B = A^T of the same S tile, the same register pair feeds SRC0/SRC1.
// D (16x16 f32, 8 VGPRs): VGPR r, lanes 0-15 -> M=r,  N=lane;
//                                  lanes 16-31 -> M=8+r, N=lane-16.
// Diagonal extracted via LDS (no dynamic vector-component indexing).
// ---------------------------------------------------------------------------
__global__ __launch_bounds__(32) void fm_finalize_kernel(
    const float* __restrict__ S,
    const float* __restrict__ T,
    float*       __restrict__ out)
{
  __shared__ float Dtile[16][16];
  const int lane = threadIdx.x;   // 0..31, exactly one wave, EXEC all ones
  const int m0   = blockIdx.x * 16;
  const int half = lane >> 4;     // 0 or 1
  const int mn   = lane & 15;     // M (for A) == N (for B)

  v8f acc = {};
#pragma unroll
  for (int c = 0; c < 4; ++c) {
    const int kbase = 4 * c + half * 2;
    v2f a;
    a.x = S[(m0 + mn) * FM_K + kbase + 0];
    a.y = S[(m0 + mn) * FM_K + kbase + 1];
    // D += A * A^T  (8 args: neg_a, A, neg_b, B, c_mod, C, reuse_a, reuse_b)
    acc = __builtin_amdgcn_wmma_f32_16x16x4_f32(
        false, a, false, a, (short)0, acc, false, false);
  }

#pragma unroll
  for (int r = 0; r < 8; ++r)
    Dtile[r + 8 * half][mn] = acc[r];
  __syncthreads();

  if (lane < 16) {
    const int m = m0 + lane;
    out[m] = Dtile[lane][lane] - T[m];
  }
}

// ---------------------------------------------------------------------------
// Launcher. Inputs in setup_inputs() order:
//   d_in[0]=indices i32[NNZ], d_in[1]=batch_ids i32[NNZ],
//   d_in[2]=values f32[NNZ],  d_in[3]=weight f32[VOCAB*K]
// d_out: f32[BATCH] (shape [4096,1] flattened).
// d_ws layout: S = BATCH*K floats, then T = BATCH floats (~278 KB).
// ---------------------------------------------------------------------------
extern "C" void kernel_launch(void* const* d_in, const int* in_sizes, int n_in,
                              void* d_out, int out_size, void* d_ws, size_t ws_size,
                              hipStream_t stream) {
  const int*   indices   = (const int*)d_in[0];
  const int*   batch_ids = (const int*)d_in[1];
  const float* values    = (const float*)d_in[2];
  const float* weight    = (const float*)d_in[3];

  float* S = (float*)d_ws;
  float* T = S + FM_BATCH * FM_K;
  float* out = (float*)d_out;

  const int accN = FM_BATCH * (FM_K + 1);
  fm_zero_kernel<<<(accN + 255) / 256, 256, 0, stream>>>(S, accN);

  fm_scatter_kernel<<<FM_NNZ / 256, 256, 0, stream>>>(
      indices, batch_ids, values, weight, S, T);

  fm_finalize_kernel<<<FM_BATCH / 16, 32, 0, stream>>>(S, T, out);
}